// MCGCSPFFNet_53635551592933
// MI455X (gfx1250) — compile-verified
//
#include <hip/hip_runtime.h>
#include <hip/hip_bf16.h>

typedef __attribute__((ext_vector_type(2))) float v2f;
typedef __attribute__((ext_vector_type(8))) float v8f;

#define NN 20000
#define NEDGE 320000
#define NTEST 4000
#define BN_EPS 1e-5f

// Guaranteed hardware float atomic (no CAS-loop fallback), device scope since
// edge scatters cross WGPs. Non-returning form -> STOREcnt; implicit
// S_WAIT_IDLE before S_ENDPGM covers completion.
__device__ __forceinline__ void atomF(float* p, float v) {
  asm volatile("global_atomic_add_f32 %0, %1, off scope:SCOPE_DEV"
               :
               : "v"(p), "v"(v)
               : "memory");
}

// ---------------- elementwise helpers ----------------
__global__ void k_zero(float* __restrict__ p, long n) {
  long i = (long)blockIdx.x * blockDim.x + threadIdx.x;
  if (i < n) p[i] = 0.f;
}
__global__ void k_eadd(const float* __restrict__ a, const float* __restrict__ b,
                       float* __restrict__ c, long n) {
  long i = (long)blockIdx.x * blockDim.x + threadIdx.x;
  if (i < n) c[i] = a[i] + b[i];
}
__global__ void k_emul(const float* __restrict__ a, const float* __restrict__ b,
                       float* __restrict__ c, long n) {
  long i = (long)blockIdx.x * blockDim.x + threadIdx.x;
  if (i < n) c[i] = a[i] * b[i];
}
// tx2 = 2*tx2 - x
__global__ void k_comb2(float* __restrict__ tx2, const float* __restrict__ x, long n) {
  long i = (long)blockIdx.x * blockDim.x + threadIdx.x;
  if (i < n) tx2[i] = 2.f * tx2[i] - x[i];
}
// c = coef[ci]*(a+b)
__global__ void k_fuse(const float* __restrict__ a, const float* __restrict__ b,
                       const float* __restrict__ coef, int ci, float* __restrict__ c, long n) {
  long i = (long)blockIdx.x * blockDim.x + threadIdx.x;
  if (i < n) c[i] = coef[ci] * (a[i] + b[i]);
}
// batchnorm eval, 64 channels
__global__ void k_bn(const float* __restrict__ x, const float* __restrict__ g,
                     const float* __restrict__ b, const float* __restrict__ m,
                     const float* __restrict__ v, float* __restrict__ y, long n) {
  long i = (long)blockIdx.x * blockDim.x + threadIdx.x;
  if (i >= n) return;
  int c = (int)(i & 63);
  y[i] = (x[i] - m[c]) * rsqrtf(v[c] + BN_EPS) * g[c] + b[c];
}

// ---------------- edge net ----------------
__global__ void k_edge_mlp(const float* __restrict__ eni, const float* __restrict__ W1,
                           const float* __restrict__ b1, const float* __restrict__ W2,
                           const float* __restrict__ b2, float* __restrict__ ew, int nE) {
  int e = blockIdx.x * blockDim.x + threadIdx.x;
  if (e >= nE) return;
  const float* row = eni + (long)e * 12;
  float in[12];
#pragma unroll
  for (int k = 0; k < 12; ++k) in[k] = row[k];
  float acc = b2[0];
#pragma unroll
  for (int j = 0; j < 6; ++j) {
    float h = b1[j];
#pragma unroll
    for (int k = 0; k < 12; ++k) h += in[k] * W1[k * 6 + j];
    h = fmaxf(h, 0.f);
    acc += h * W2[j];
  }
  ew[e] = 1.f / (1.f + expf(-acc));
}

__global__ void k_scatter_deg(const int* __restrict__ src, const float* __restrict__ ew,
                              float* __restrict__ deg, int nE) {
  int e = blockIdx.x * blockDim.x + threadIdx.x;
  if (e >= nE) return;
  atomF(&deg[src[e]], ew[e]);
}

__global__ void k_edge_norm(const int* __restrict__ src, const int* __restrict__ dst,
                            const float* __restrict__ ew, const float* __restrict__ deg,
                            float* __restrict__ norm, int nE) {
  int e = blockIdx.x * blockDim.x + threadIdx.x;
  if (e >= nE) return;
  float ds = deg[src[e]], dd = deg[dst[e]];
  float a = ds > 0.f ? 1.f / sqrtf(ds) : 0.f;
  float b = dd > 0.f ? 1.f / sqrtf(dd) : 0.f;
  norm[e] = -a * ew[e] * b;
}

// out[dst[e], :] += norm[e]*x[src[e], :]   (sh = log2(cin/4) float4 groups per edge)
__global__ void k_prop(const int* __restrict__ src, const int* __restrict__ dst,
                       const float* __restrict__ norm, const float* __restrict__ x,
                       float* __restrict__ out, int nE, int cin, int sh) {
  long i = (long)blockIdx.x * blockDim.x + threadIdx.x;
  long total = (long)nE << sh;
  if (i >= total) return;
  int e = (int)(i >> sh);
  int c4 = (int)(i & ((1 << sh) - 1)) << 2;
  float w = norm[e];
  int s = src[e], d = dst[e];
  float4 xv = *(const float4*)(x + (long)s * cin + c4);
  float* op = out + (long)d * cin + c4;
  atomF(op + 0, w * xv.x);
  atomF(op + 1, w * xv.y);
  atomF(op + 2, w * xv.z);
  atomF(op + 3, w * xv.w);
}

// ---------------- WMMA f32 16x16x4 matmuls ----------------
// out = relu( X0@W[0] + X1@W[1] + X2@W[2] + bias ),  W: (3, cin, 64) row-major
__global__ __launch_bounds__(256) void k_cheb_wmma(
    const float* __restrict__ X0, const float* __restrict__ X1, const float* __restrict__ X2,
    const float* __restrict__ W, const float* __restrict__ bias,
    float* __restrict__ out, int nrows, int cin) {
  int wave = threadIdx.x >> 5, lane = threadIdx.x & 31;
  int tile = blockIdx.x * 8 + wave;
  int ntiles = nrows >> 4;
  if (tile >= ntiles) return;
  int row0 = tile << 4;
  int midx = lane & 15;
  int hi = lane >> 4;  // 0: K-pair (0,1)   1: K-pair (2,3)
  v8f acc0 = {0,0,0,0,0,0,0,0}, acc1 = {0,0,0,0,0,0,0,0};
  v8f acc2 = {0,0,0,0,0,0,0,0}, acc3 = {0,0,0,0,0,0,0,0};
#pragma unroll
  for (int m = 0; m < 3; ++m) {
    const float* Xm = (m == 0) ? X0 : (m == 1) ? X1 : X2;
    const float* Wm = W + (long)m * cin * 64;
    const float* arow = Xm + (long)(row0 + midx) * cin + 2 * hi;
    for (int k = 0; k < cin; k += 4) {
      float2 av = *(const float2*)(arow + k);
      v2f a = {av.x, av.y};
      const float* wk0 = Wm + (long)(k + 2 * hi) * 64 + midx;
      const float* wk1 = wk0 + 64;
      v2f b0 = {wk0[0],  wk1[0]};
      v2f b1 = {wk0[16], wk1[16]};
      v2f b2 = {wk0[32], wk1[32]};
      v2f b3 = {wk0[48], wk1[48]};
      acc0 = __builtin_amdgcn_wmma_f32_16x16x4_f32(false, a, false, b0, (short)0, acc0, false, false);
      acc1 = __builtin_amdgcn_wmma_f32_16x16x4_f32(false, a, false, b1, (short)0, acc1, false, false);
      acc2 = __builtin_amdgcn_wmma_f32_16x16x4_f32(false, a, false, b2, (short)0, acc2, false, false);
      acc3 = __builtin_amdgcn_wmma_f32_16x16x4_f32(false, a, false, b3, (short)0, acc3, false, false);
    }
  }
  int rbase = row0 + hi * 8;
  v8f accs[4] = {acc0, acc1, acc2, acc3};
#pragma unroll
  for (int ct = 0; ct < 4; ++ct) {
    int col = ct * 16 + midx;
    float bv = bias[col];
#pragma unroll
    for (int r = 0; r < 8; ++r) {
      float v = accs[ct][r] + bv;
      out[(long)(rbase + r) * 64 + col] = fmaxf(v, 0.f);
    }
  }
}

// out = epilogue( X@W + bias );  mode 0: none, 2: batchnorm
__global__ __launch_bounds__(256) void k_linear64_wmma(
    const float* __restrict__ X, const float* __restrict__ W, const float* __restrict__ bias,
    float* __restrict__ out, int nrows, int cin, int mode,
    const float* __restrict__ bng, const float* __restrict__ bnb,
    const float* __restrict__ bnm, const float* __restrict__ bnv) {
  int wave = threadIdx.x >> 5, lane = threadIdx.x & 31;
  int tile = blockIdx.x * 8 + wave;
  int ntiles = nrows >> 4;
  if (tile >= ntiles) return;
  int row0 = tile << 4;
  int midx = lane & 15;
  int hi = lane >> 4;
  v8f acc0 = {0,0,0,0,0,0,0,0}, acc1 = {0,0,0,0,0,0,0,0};
  v8f acc2 = {0,0,0,0,0,0,0,0}, acc3 = {0,0,0,0,0,0,0,0};
  const float* arow = X + (long)(row0 + midx) * cin + 2 * hi;
  for (int k = 0; k < cin; k += 4) {
    float2 av = *(const float2*)(arow + k);
    v2f a = {av.x, av.y};
    const float* wk0 = W + (long)(k + 2 * hi) * 64 + midx;
    const float* wk1 = wk0 + 64;
    v2f b0 = {wk0[0],  wk1[0]};
    v2f b1 = {wk0[16], wk1[16]};
    v2f b2 = {wk0[32], wk1[32]};
    v2f b3 = {wk0[48], wk1[48]};
    acc0 = __builtin_amdgcn_wmma_f32_16x16x4_f32(false, a, false, b0, (short)0, acc0, false, false);
    acc1 = __builtin_amdgcn_wmma_f32_16x16x4_f32(false, a, false, b1, (short)0, acc1, false, false);
    acc2 = __builtin_amdgcn_wmma_f32_16x16x4_f32(false, a, false, b2, (short)0, acc2, false, false);
    acc3 = __builtin_amdgcn_wmma_f32_16x16x4_f32(false, a, false, b3, (short)0, acc3, false, false);
  }
  int rbase = row0 + hi * 8;
  v8f accs[4] = {acc0, acc1, acc2, acc3};
#pragma unroll
  for (int ct = 0; ct < 4; ++ct) {
    int col = ct * 16 + midx;
    float bv = bias[col];
    float sc = 1.f, sh2 = 0.f;
    if (mode == 2) {
      sc = rsqrtf(bnv[col] + BN_EPS) * bng[col];
      sh2 = bnb[col] - bnm[col] * sc;
    }
#pragma unroll
    for (int r = 0; r < 8; ++r) {
      float v = accs[ct][r] + bv;
      if (mode == 2) v = v * sc + sh2;
      out[(long)(rbase + r) * 64 + col] = v;
    }
  }
}

// 2-class head: out[n,:] = softmax(h[n,:]@W2 + b2)
__global__ void k_head2(const float* __restrict__ H, const float* __restrict__ W2,
                        const float* __restrict__ b2, float* __restrict__ out, int n) {
  int i = blockIdx.x * blockDim.x + threadIdx.x;
  if (i >= n) return;
  const float* h = H + (long)i * 64;
  float p0 = b2[0], p1 = b2[1];
#pragma unroll
  for (int c = 0; c < 64; ++c) {
    float hv = h[c];
    p0 += hv * W2[c * 2];
    p1 += hv * W2[c * 2 + 1];
  }
  float mx = fmaxf(p0, p1);
  float e0 = expf(p0 - mx), e1 = expf(p1 - mx);
  float s = e0 + e1;
  out[(long)i * 2] = e0 / s;
  out[(long)i * 2 + 1] = e1 / s;
}

// accuracy counting over test indices, 6 predictions living in d_out
__global__ void k_acc(const int* __restrict__ ind, const int* __restrict__ labels,
                      const float* __restrict__ preds, int* __restrict__ cnt) {
  int t = blockIdx.x * blockDim.x + threadIdx.x;
  if (t >= NTEST * 6) return;
  int k = t / NTEST, tt = t - k * NTEST;
  int idx = ind[tt];
  const float* p = preds + (long)k * (2 * NN) + (long)idx * 2;
  int am = (p[1] > p[0]) ? 1 : 0;
  if (am == labels[idx]) atomicAdd(&cnt[k], 1);
}

__global__ void k_coef(const int* __restrict__ cnt, float* __restrict__ coef) {
  if (blockIdx.x == 0 && threadIdx.x == 0) {
    float w[6], sw = 0.f;
    for (int k = 0; k < 6; ++k) { w[k] = (float)cnt[k] / (float)NTEST; sw += w[k]; }
    coef[0] = (w[0] + w[1]) / sw;
    coef[1] = (w[2] + w[3]) / sw;
    coef[2] = (w[4] + w[5]) / sw;
  }
}

// per-row softmax(Q*K) * (Va+Vb) + X -> concat; one wave per row, 2 channels/lane
__global__ __launch_bounds__(256) void k_attn(
    const float* __restrict__ S1, const float* __restrict__ S2, const float* __restrict__ S3,
    const float* __restrict__ V1, const float* __restrict__ V2, const float* __restrict__ V3,
    const float* __restrict__ X1, const float* __restrict__ X2, const float* __restrict__ X3,
    float* __restrict__ attcat, int n) {
  int wave = threadIdx.x >> 5, lane = threadIdx.x & 31;
  int row = blockIdx.x * 8 + wave;
  if (row >= n) return;
  long base = (long)row * 64;
  const float* Ss[3] = {S1, S2, S3};
  const float* Va[3] = {V2, V1, V1};
  const float* Vb[3] = {V3, V3, V2};
  const float* Xs[3] = {X1, X2, X3};
#pragma unroll
  for (int i = 0; i < 3; ++i) {
    float s0 = Ss[i][base + lane], s1 = Ss[i][base + 32 + lane];
    float mx = fmaxf(s0, s1);
    for (int o = 16; o > 0; o >>= 1) mx = fmaxf(mx, __shfl_xor(mx, o, 32));
    float e0 = expf(s0 - mx), e1 = expf(s1 - mx);
    float sum = e0 + e1;
    for (int o = 16; o > 0; o >>= 1) sum += __shfl_xor(sum, o, 32);
    float inv = 1.f / sum;
    float a0 = e0 * inv * (Va[i][base + lane] + Vb[i][base + lane]) + Xs[i][base + lane];
    float a1 = e1 * inv * (Va[i][base + 32 + lane] + Vb[i][base + 32 + lane]) + Xs[i][base + 32 + lane];
    attcat[(long)row * 192 + i * 64 + lane] = a0;
    attcat[(long)row * 192 + i * 64 + 32 + lane] = a1;
  }
}

// ---------------- host ----------------
extern "C" void kernel_launch(void* const* d_in, const int* in_sizes, int n_in,
                              void* d_out, int out_size, void* d_ws, size_t ws_size,
                              hipStream_t stream) {
  const int* ind    = (const int*)d_in[0];
  const int* labels = (const int*)d_in[1];
  const float* f1 = (const float*)d_in[2];
  const float* f2 = (const float*)d_in[3];
  const float* f3 = (const float*)d_in[4];
  const int* ei1 = (const int*)d_in[5];  const float* en1 = (const float*)d_in[6];
  const int* ei2 = (const int*)d_in[7];  const float* en2 = (const float*)d_in[8];
  const int* ei3 = (const int*)d_in[9];  const float* en3 = (const float*)d_in[10];

  // params: JAX pytree flatten order (dict keys sorted, recursive): att, cls, g1, g2, g3, mlp
  static const int leaf_sizes[72] = {
      64,64,64,64, 64,64,64,64, 64,64,64,64,                               // att bn1..bn3 (b,g,m,v)
      4096,64, 4096,64, 4096,64,                                           // k1,k2,k3 (W,b)
      4096,64, 4096,64, 4096,64,                                           // q1,q2,q3
      4096,64, 4096,64, 4096,64,                                           // v1,v2,v3
      12288,128,64,2,                                                      // cls W1,W2,b1,b2
      24576,64,12288,64,12288,64,72,6,6,1,                                 // g1
      24576,64,12288,64,12288,64,72,6,6,1,                                 // g2
      24576,64,12288,64,12288,64,72,6,6,1,                                 // g3
      4096,128,64,2,64,64,64,64};                                          // mlp W1,W2,b1,b2,bn_b,bn_g,bn_m,bn_v
  const float* P[72];
  if (n_in >= 11 + 72) {
    for (int i = 0; i < 72; ++i) P[i] = (const float*)d_in[11 + i];
  } else {
    const float* base = (const float*)d_in[11];
    long off = 0;
    for (int i = 0; i < 72; ++i) { P[i] = base + off; off += leaf_sizes[i]; }
  }
  // attention params
  const float *bnG[3] = {P[1], P[5], P[9]}, *bnB[3] = {P[0], P[4], P[8]};
  const float *bnM[3] = {P[2], P[6], P[10]}, *bnV[3] = {P[3], P[7], P[11]};
  const float *kW[3] = {P[12], P[14], P[16]}, *kB[3] = {P[13], P[15], P[17]};
  const float *qW[3] = {P[18], P[20], P[22]}, *qB[3] = {P[19], P[21], P[23]};
  const float *vW[3] = {P[24], P[26], P[28]}, *vB[3] = {P[25], P[27], P[29]};
  const float *clsW1 = P[30], *clsW2 = P[31], *clsB1 = P[32], *clsB2 = P[33];
  struct GP { const float *c0W,*c0b,*c1W,*c1b,*c2W,*c2b,*eW1,*eW2,*eb1,*eb2; };
  GP g[3];
  for (int gi = 0; gi < 3; ++gi) {
    const float** q = P + 34 + gi * 10;
    g[gi] = GP{q[0], q[1], q[2], q[3], q[4], q[5], q[6], q[7], q[8], q[9]};
  }
  const float *mlpW1 = P[64], *mlpW2 = P[65], *mlpB1 = P[66], *mlpB2 = P[67];
  const float *mlpBnB = P[68], *mlpBnG = P[69], *mlpBnM = P[70], *mlpBnV = P[71];

  // workspace layout
  float* wsf = (float*)d_ws;
  float* ew   = wsf;
  float* nrm  = ew + NEDGE;
  float* deg  = nrm + NEDGE;
  int*   cnt  = (int*)(deg + NN);
  float* coef = (float*)(cnt + 8);
  float* pool = coef + 8;
  float* tx1 = pool;                       // NN*128
  float* tx2 = tx1 + (long)NN * 128;       // NN*128
  float* h0  = tx2 + (long)NN * 128;       // NN*64
  float* h1  = h0 + (long)NN * 64;         // NN*64
  float* xt  = h1 + (long)NN * 64;         // NN*64
  float* ob[6];
  for (int k = 0; k < 6; ++k) ob[k] = xt + (long)NN * 64 + (long)k * NN * 64;
  float* cb[3];
  for (int k = 0; k < 3; ++k) cb[k] = ob[0] + (long)6 * NN * 64 + (long)k * NN * 64;
  // phase aliases
  float* hbuf = h0;                                    // mlp/cls hidden
  float* bnx[3] = {tx1, tx1 + (long)NN * 64, tx1 + (long)NN * 128};
  float* tmpQ = xt; float* tmpK = h1;
  float* Sb[3] = {ob[0], ob[1], ob[2]};
  float* Vb2[3] = {ob[3], ob[4], ob[5]};
  float* attcat = cb[0];                               // NN*192

  float* outp = (float*)d_out;
  const int TB = 256;
  auto G = [](long n) { return (unsigned)((n + 255) / 256); };
  auto zero = [&](float* p, long n) { k_zero<<<G(n), TB, 0, stream>>>(p, n); };
  const unsigned mmBlocks = (unsigned)(((NN / 16) + 7) / 8);  // 157

  auto chebLayer = [&](const float* xin, int cin, const float* W, const float* bias,
                       float* out, const int* src, const int* dst) {
    int sh = (cin == 128) ? 5 : 4;
    zero(tx1, (long)NN * cin);
    k_prop<<<G((long)NEDGE << sh), TB, 0, stream>>>(src, dst, nrm, xin, tx1, NEDGE, cin, sh);
    zero(tx2, (long)NN * cin);
    k_prop<<<G((long)NEDGE << sh), TB, 0, stream>>>(src, dst, nrm, tx1, tx2, NEDGE, cin, sh);
    k_comb2<<<G((long)NN * cin), TB, 0, stream>>>(tx2, xin, (long)NN * cin);
    k_cheb_wmma<<<mmBlocks, TB, 0, stream>>>(xin, tx1, tx2, W, bias, out, NN, cin);
  };

  auto aelngcn = [&](const GP& gp, const float* feat, const int* ei, const float* eni, float* o) {
    const int* src = ei; const int* dst = ei + NEDGE;
    k_edge_mlp<<<G(NEDGE), TB, 0, stream>>>(eni, gp.eW1, gp.eb1, gp.eW2, gp.eb2, ew, NEDGE);
    zero(deg, NN);
    k_scatter_deg<<<G(NEDGE), TB, 0, stream>>>(src, ew, deg, NEDGE);
    k_edge_norm<<<G(NEDGE), TB, 0, stream>>>(src, dst, ew, deg, nrm, NEDGE);
    chebLayer(feat, 128, gp.c0W, gp.c0b, h0, src, dst);
    chebLayer(h0, 64, gp.c1W, gp.c1b, h1, src, dst);
    k_eadd<<<G((long)NN * 64), TB, 0, stream>>>(h0, h1, xt, (long)NN * 64);
    chebLayer(xt, 64, gp.c2W, gp.c2b, o, src, dst);
  };

  // six graph branches
  aelngcn(g[0], f1, ei3, en3, ob[0]);
  aelngcn(g[0], f1, ei2, en2, ob[1]);
  aelngcn(g[1], f2, ei1, en1, ob[2]);
  aelngcn(g[1], f2, ei3, en3, ob[3]);
  aelngcn(g[2], f3, ei2, en2, ob[4]);
  aelngcn(g[2], f3, ei1, en1, ob[5]);

  auto mlpHead = [&](const float* x, float* dst) {
    k_linear64_wmma<<<mmBlocks, TB, 0, stream>>>(x, mlpW1, mlpB1, hbuf, NN, 64, 2,
                                                 mlpBnG, mlpBnB, mlpBnM, mlpBnV);
    k_head2<<<G(NN), TB, 0, stream>>>(hbuf, mlpW2, mlpB2, dst, NN);
  };
  for (int k = 0; k < 6; ++k) mlpHead(ob[k], outp + (long)(1 + k) * 2 * NN);

  // accuracy-based fusion weights
  zero((float*)cnt, 8);
  k_acc<<<G((long)NTEST * 6), TB, 0, stream>>>(ind, labels, outp + (long)2 * NN, cnt);
  k_coef<<<1, 32, 0, stream>>>(cnt, coef);
  for (int i = 0; i < 3; ++i)
    k_fuse<<<G((long)NN * 64), TB, 0, stream>>>(ob[2 * i], ob[2 * i + 1], coef, i, cb[i],
                                                (long)NN * 64);
  for (int i = 0; i < 3; ++i) mlpHead(cb[i], outp + (long)(7 + i) * 2 * NN);

  // attention fusion
  for (int i = 0; i < 3; ++i)
    k_bn<<<G((long)NN * 64), TB, 0, stream>>>(cb[i], bnG[i], bnB[i], bnM[i], bnV[i],
                                              bnx[i], (long)NN * 64);
  for (int i = 0; i < 3; ++i) {
    k_linear64_wmma<<<mmBlocks, TB, 0, stream>>>(bnx[i], qW[i], qB[i], tmpQ, NN, 64, 0,
                                                 nullptr, nullptr, nullptr, nullptr);
    k_linear64_wmma<<<mmBlocks, TB, 0, stream>>>(bnx[i], kW[i], kB[i], tmpK, NN, 64, 0,
                                                 nullptr, nullptr, nullptr, nullptr);
    k_emul<<<G((long)NN * 64), TB, 0, stream>>>(tmpQ, tmpK, Sb[i], (long)NN * 64);
    k_linear64_wmma<<<mmBlocks, TB, 0, stream>>>(bnx[i], vW[i], vB[i], Vb2[i], NN, 64, 0,
                                                 nullptr, nullptr, nullptr, nullptr);
  }
  k_attn<<<(NN + 7) / 8, TB, 0, stream>>>(Sb[0], Sb[1], Sb[2], Vb2[0], Vb2[1], Vb2[2],
                                          bnx[0], bnx[1], bnx[2], attcat, NN);
  // classifier
  k_linear64_wmma<<<mmBlocks, TB, 0, stream>>>(attcat, clsW1, clsB1, hbuf, NN, 192, 0,
                                               nullptr, nullptr, nullptr, nullptr);
  k_head2<<<G(NN), TB, 0, stream>>>(hbuf, clsW2, clsB2, outp, NN);
}